// GroupedGemmRunner_44848048505297
// MI455X (gfx1250) — compile-verified
//
#include <hip/hip_runtime.h>
#include <math.h>

typedef __attribute__((ext_vector_type(2))) float v2f;
typedef __attribute__((ext_vector_type(8))) float v8f;
typedef __attribute__((ext_vector_type(4))) int   v4i;

#define BM 128
#define BN 128
#define BK 32
#define LSTR 36   // LDS row stride in floats: 144B -> 16B-aligned b128 stores, no bank conflicts

// Stage 16 bytes global -> LDS. Prefer the gfx1250 async-to-LDS path (ASYNCcnt),
// fall back to a register round-trip if the builtin is absent on this toolchain.
__device__ __forceinline__ void stage16(const float* __restrict__ g, float* l) {
#if defined(__AMDGCN__) && __has_builtin(__builtin_amdgcn_global_load_async_to_lds_b128)
  __builtin_amdgcn_global_load_async_to_lds_b128(
      (__attribute__((address_space(1))) v4i*)g,
      (__attribute__((address_space(3))) v4i*)l, 0, 0);
#else
  *(float4*)l = *(const float4*)g;
#endif
}

__device__ __forceinline__ void wait_stage() {
#if defined(__AMDGCN__) && __has_builtin(__builtin_amdgcn_s_wait_asynccnt)
  __builtin_amdgcn_s_wait_asynccnt(0);
#elif defined(__AMDGCN__)
  asm volatile("s_wait_asynccnt 0" ::: "memory");
#endif
}

__global__ __launch_bounds__(256) void grouped_gemm_f32_wmma(
    const float* __restrict__ A,      // [M, K]
    const float* __restrict__ B,      // [E, N, K] (column-major weights)
    float* __restrict__ C,            // [M, N]
    const int* __restrict__ segp,     // [E+1]
    const int* __restrict__ widxp,    // [E]
    int M, int N, int K, int E) {
  __shared__ float sA[BM * LSTR];
  __shared__ float sB[BN * LSTR];

  const int tid  = threadIdx.x;
  const int wave = tid >> 5;
  const int lane = tid & 31;
  const int l15  = lane & 15;
  const int half = lane >> 4;

  const int rowStart = blockIdx.y * BM;
  const int colStart = blockIdx.x * BN;

  for (int s = 0; s < E; ++s) {
    const int lo = segp[s];
    const int hi = segp[s + 1];
    const int segLo = lo > rowStart ? lo : rowStart;
    const int segHi = hi < rowStart + BM ? hi : rowStart + BM;
    if (segLo >= segHi) continue;   // uniform across the block (scalar branch)

    const int widx = widxp[s];
    const float* __restrict__ Bexp = B + (size_t)widx * (size_t)N * (size_t)K;

    v8f zero = {};
    v8f acc[8];
#pragma unroll
    for (int i = 0; i < 8; ++i) acc[i] = zero;

    for (int kb = 0; kb < K; kb += BK) {
      __syncthreads();  // protect LDS from previous iteration's readers
      // 2048 float4 chunks per tile pair; 4 (A) + 4 (B) per thread
#pragma unroll
      for (int i = 0; i < 4; ++i) {
        const int ch = tid + i * 256;
        const int r  = ch >> 3;          // 0..127 (tile row)
        const int cc = (ch & 7) << 2;    // 0,4,...,28 (k offset in floats)
        stage16(A + (size_t)(rowStart + r) * K + kb + cc, &sA[r * LSTR + cc]);
        stage16(Bexp + (size_t)(colStart + r) * K + kb + cc, &sB[r * LSTR + cc]);
      }
      wait_stage();
      __syncthreads();

      const int aRow = wave * 16 + l15;
#pragma unroll
      for (int kk = 0; kk < 8; ++kk) {
        // A 16x4 layout: lanes 0-15 hold K={0,1}, lanes 16-31 hold K={2,3}
        const int kbase = (kk << 2) + (half << 1);
        const v2f aF = *(const v2f*)&sA[aRow * LSTR + kbase];
#pragma unroll
        for (int nt = 0; nt < 8; ++nt) {
          const v2f bF = *(const v2f*)&sB[(nt * 16 + l15) * LSTR + kbase];
          acc[nt] = __builtin_amdgcn_wmma_f32_16x16x4_f32(
              false, aF, false, bF, (short)0, acc[nt], false, false);
        }
      }
    }

    // C/D 16x16 f32 layout: VGPR v, lanes 0-15 -> M=v, lanes 16-31 -> M=v+8
    const int rowBase = rowStart + wave * 16 + (half << 3);
    const int colB = colStart + l15;
#pragma unroll
    for (int nt = 0; nt < 8; ++nt) {
      const int col = colB + nt * 16;
#pragma unroll
      for (int v = 0; v < 8; ++v) {
        const int row = rowBase + v;
        if (row >= segLo && row < segHi)
          C[(size_t)row * N + col] = acc[nt][v];
      }
    }
  }
}

extern "C" void kernel_launch(void* const* d_in, const int* in_sizes, int n_in,
                              void* d_out, int out_size, void* d_ws, size_t ws_size,
                              hipStream_t stream) {
  // setup_inputs order: a, b, c, batch_size, weight_column_major, seg_indptr, weight_indices
  const float* a = (const float*)d_in[0];
  const float* b = (const float*)d_in[1];
  const int* seg_indptr = (const int*)d_in[5];
  const int* weight_indices = (const int*)d_in[6];

  const int E = in_sizes[6];
  const long long MK = in_sizes[0];
  const long long CN = out_size;
  const long long NK = in_sizes[1] / E;
  const int M = (int)(sqrt((double)MK * (double)CN / (double)NK) + 0.5);
  const int K = (int)(MK / M);
  const int N = (int)(CN / M);

  dim3 grid(N / BN, M / BM);
  dim3 block(256);
  grouped_gemm_f32_wmma<<<grid, block, 0, stream>>>(
      a, b, (float*)d_out, seg_indptr, weight_indices, M, N, K, E);
}